// GIN_Node_With_Edge_Encoder_266287972763
// MI455X (gfx1250) — compile-verified
//
#include <hip/hip_runtime.h>
#include <hip/hip_bf16.h>

// ---------------------------------------------------------------------------
// GIN (2 layers) : edge scatter-add  ->  fused MLP (WMMA f32 16x16x4)
//                  -> BN stats fused in epilogue -> BN apply
// ---------------------------------------------------------------------------

typedef __attribute__((ext_vector_type(2))) float v2f;
typedef __attribute__((ext_vector_type(8))) float v8f;

#define DIM   128
#define HID2  256
#define BN_EPS 1e-5f

// --------------------------- zero fill -------------------------------------
__global__ void zero_f32(float* __restrict__ p, int n) {
    int i = blockIdx.x * blockDim.x + threadIdx.x;
    int stride = gridDim.x * blockDim.x;
    for (; i < n; i += stride) p[i] = 0.0f;
}

// --------------------------- edge scatter ----------------------------------
// one wave (32 lanes) per edge; each lane owns 4 consecutive features.
// gather x[src] (float4), scale by edge weight, atomic-add into agg[dst].
__global__ void edge_scatter(const float* __restrict__ x,
                             const int*   __restrict__ src,
                             const int*   __restrict__ dst,
                             const float* __restrict__ ew,
                             float*       __restrict__ agg,
                             int nEdges) {
    long long gid = (long long)blockIdx.x * blockDim.x + threadIdx.x;
    long long total = (long long)nEdges * 32;
    if (gid >= total) return;
    int e  = (int)(gid >> 5);
    int c4 = ((int)gid & 31) * 4;
    int s = src[e];
    int d = dst[e];
    float w = ew[e];
    const float4 v = *reinterpret_cast<const float4*>(x + (size_t)s * DIM + c4);
    float* o = agg + (size_t)d * DIM + c4;
    atomicAdd(o + 0, v.x * w);
    atomicAdd(o + 1, v.y * w);
    atomicAdd(o + 2, v.z * w);
    atomicAdd(o + 3, v.w * w);
}

// --------------------------- fused GIN MLP ---------------------------------
// block = 256 threads (8 waves), one block per 16 node rows.
//   h = agg + x                                  (staged to LDS, 16x128)
//   t1 = relu(h @ Wa + ba)                        (LDS, 16x256)  GEMM1: WMMA
//   out = relu(t1 @ Wb + bb)                      (global)       GEMM2: WMMA
//   per-column sum / sum^2 reduced via LDS ds_add_f32, then global atomics.
template <int OUTC>
__global__ __launch_bounds__(256)
void gin_mlp(const float* __restrict__ agg,   // [N,128]
             const float* __restrict__ xin,   // [N,128]
             const float* __restrict__ Wa,    // [128,256]
             const float* __restrict__ ba,    // [256]
             const float* __restrict__ Wb,    // [256,OUTC]
             const float* __restrict__ bb,    // [OUTC]
             float*       __restrict__ out,   // [N,OUTC]
             float*       __restrict__ statSum,
             float*       __restrict__ statSq,
             int nNodes) {
    __shared__ float Ablk[16 * DIM];    // 8 KB : h = agg + x
    __shared__ float T1[16 * HID2];     // 16 KB : hidden activations
    __shared__ float colSum[DIM];
    __shared__ float colSq[DIM];

    const int tid   = threadIdx.x;
    const int wave  = tid >> 5;
    const int lane  = tid & 31;
    const int l16   = lane & 15;
    const int khalf = (lane >> 4) << 1;     // 0 for lanes 0-15, 2 for 16-31
    const int rbase = (lane >> 4) << 3;     // C/D rows 0..7 or 8..15
    const int m0    = blockIdx.x * 16;

    if (tid < OUTC) { colSum[tid] = 0.0f; colSq[tid] = 0.0f; }

    // stage A block (h = agg + x), 2048 elems, 8 per thread
    for (int i = tid; i < 16 * DIM; i += 256) {
        int r = i >> 7;
        size_t g = (size_t)(m0 + r) * DIM + (i & 127);
        Ablk[i] = (m0 + r < nNodes) ? (agg[g] + xin[g]) : 0.0f;
    }
    __syncthreads();

    // ---- GEMM1: each wave -> 2 column tiles (32 of the 256 hidden cols) ----
    {
        v8f acc0 = {};
        v8f acc1 = {};
        const int c0 = wave * 32 + l16;
        const int c1 = c0 + 16;
        for (int k = 0; k < DIM; k += 4) {
            const int kk = k + khalf;
            v2f a;  a.x  = Ablk[l16 * DIM + kk];
                    a.y  = Ablk[l16 * DIM + kk + 1];
            v2f b0; b0.x = Wa[kk * HID2 + c0];
                    b0.y = Wa[(kk + 1) * HID2 + c0];
            v2f b1; b1.x = Wa[kk * HID2 + c1];
                    b1.y = Wa[(kk + 1) * HID2 + c1];
            acc0 = __builtin_amdgcn_wmma_f32_16x16x4_f32(false, a, false, b0,
                                                         (short)0, acc0, false, false);
            acc1 = __builtin_amdgcn_wmma_f32_16x16x4_f32(false, a, false, b1,
                                                         (short)0, acc1, false, false);
        }
        const float bias0 = ba[c0];
        const float bias1 = ba[c1];
#pragma unroll
        for (int r = 0; r < 8; ++r) {
            const int row = rbase + r;
            T1[row * HID2 + c0] = fmaxf(acc0[r] + bias0, 0.0f);
            T1[row * HID2 + c1] = fmaxf(acc1[r] + bias1, 0.0f);
        }
    }
    __syncthreads();

    // ---- GEMM2: 256 -> OUTC. OUTC==128: 8 waves x 16 cols. OUTC==2: wave 0 only.
    const int nColTiles = (OUTC >= 16) ? (OUTC / 16) : 1;
    if (wave < nColTiles) {
        const int c = wave * 16 + l16;
        const bool cOk = (c < OUTC);
        v8f acc = {};
        for (int k = 0; k < HID2; k += 4) {
            const int kk = k + khalf;
            v2f a; a.x = T1[l16 * HID2 + kk];
                   a.y = T1[l16 * HID2 + kk + 1];
            v2f b; b.x = cOk ? Wb[kk * OUTC + c] : 0.0f;
                   b.y = cOk ? Wb[(kk + 1) * OUTC + c] : 0.0f;
            acc = __builtin_amdgcn_wmma_f32_16x16x4_f32(false, a, false, b,
                                                        (short)0, acc, false, false);
        }
        const float bias = cOk ? bb[c] : 0.0f;
        float lsum = 0.0f, lsq = 0.0f;
#pragma unroll
        for (int r = 0; r < 8; ++r) {
            const int row = rbase + r;
            const float v = fmaxf(acc[r] + bias, 0.0f);
            if (cOk && (m0 + row < nNodes)) {
                out[(size_t)(m0 + row) * OUTC + c] = v;
                lsum += v;
                lsq  += v * v;
            }
        }
        if (cOk) {
            atomicAdd(&colSum[c], lsum);   // LDS float atomic -> ds_add_f32
            atomicAdd(&colSq[c],  lsq);
        }
    }
    __syncthreads();
    if (tid < OUTC) {
        atomicAdd(&statSum[tid], colSum[tid]);
        atomicAdd(&statSq[tid],  colSq[tid]);
    }
}

// --------------------------- batchnorm apply -------------------------------
__global__ void bn_apply(const float* __restrict__ in,
                         float*       __restrict__ out,
                         const float* __restrict__ sum,
                         const float* __restrict__ sq,
                         const float* __restrict__ gamma,
                         const float* __restrict__ beta,
                         int nNodes, int C) {
    long long i = (long long)blockIdx.x * blockDim.x + threadIdx.x;
    long long n = (long long)nNodes * C;
    long long stride = (long long)gridDim.x * blockDim.x;
    const float invN = 1.0f / (float)nNodes;
    for (; i < n; i += stride) {
        int c = (int)(i % C);
        float mean = sum[c] * invN;
        float var  = sq[c] * invN - mean * mean;
        out[i] = (in[i] - mean) * rsqrtf(var + BN_EPS) * gamma[c] + beta[c];
    }
}

// ---------------------------------------------------------------------------
extern "C" void kernel_launch(void* const* d_in, const int* in_sizes, int n_in,
                              void* d_out, int out_size, void* d_ws, size_t ws_size,
                              hipStream_t stream) {
    const float* x   = (const float*)d_in[0];
    const int*   ei  = (const int*)d_in[1];
    // d_in[2] = edge_attr (unused by reference)
    const float* ew  = (const float*)d_in[3];
    const float* W1a = (const float*)d_in[4];
    const float* b1a = (const float*)d_in[5];
    const float* W1b = (const float*)d_in[6];
    const float* b1b = (const float*)d_in[7];
    const float* g1  = (const float*)d_in[8];
    const float* be1 = (const float*)d_in[9];
    const float* W2a = (const float*)d_in[10];
    const float* b2a = (const float*)d_in[11];
    const float* W2b = (const float*)d_in[12];
    const float* b2b = (const float*)d_in[13];
    const float* g2  = (const float*)d_in[14];
    const float* be2 = (const float*)d_in[15];

    const int N = in_sizes[0] / DIM;       // 50000
    const int E = in_sizes[3];             // 1600000
    const int* src = ei;
    const int* dst = ei + E;

    float* buf0  = (float*)d_ws;                    // [N,128] agg (reused)
    float* buf1  = buf0 + (size_t)N * DIM;          // [N,128] h after layer 1
    float* buf2  = buf1 + (size_t)N * DIM;          // [N,2]   pre-BN layer 2
    float* stats = buf2 + (size_t)N * 2;            // 512 floats of stats
    float* s1sum = stats;          // [128]
    float* s1sq  = stats + 128;    // [128]
    float* s2sum = stats + 256;    // [2]
    float* s2sq  = stats + 258;    // [2]

    const long long scatterThreads = (long long)E * 32;
    const int scatterBlocks = (int)((scatterThreads + 255) / 256);
    const int mlpBlocks = (N + 15) / 16;

    // -------- layer 1 --------
    zero_f32<<<2048, 256, 0, stream>>>(buf0, N * DIM);
    zero_f32<<<2, 256, 0, stream>>>(stats, 512);
    edge_scatter<<<scatterBlocks, 256, 0, stream>>>(x, src, dst, ew, buf0, E);
    gin_mlp<DIM><<<mlpBlocks, 256, 0, stream>>>(buf0, x, W1a, b1a, W1b, b1b,
                                                buf1, s1sum, s1sq, N);
    bn_apply<<<2048, 256, 0, stream>>>(buf1, buf1, s1sum, s1sq, g1, be1, N, DIM);

    // -------- layer 2 --------
    zero_f32<<<2048, 256, 0, stream>>>(buf0, N * DIM);
    edge_scatter<<<scatterBlocks, 256, 0, stream>>>(buf1, src, dst, ew, buf0, E);
    gin_mlp<2><<<mlpBlocks, 256, 0, stream>>>(buf0, buf1, W2a, b2a, W2b, b2b,
                                              buf2, s2sum, s2sq, N);
    bn_apply<<<256, 256, 0, stream>>>(buf2, (float*)d_out, s2sum, s2sq,
                                      g2, be2, N, 2);
}